// EvolutionBank_76836964926208
// MI455X (gfx1250) — compile-verified
//
#include <hip/hip_runtime.h>

// EvolutionBank: scatter emb into circular bank slot, then temporal-consistency
// over the 6x32 window. Memory-bound (~182 MB / launch): stream gathered bank
// rows to LDS via CDNA5 async global->LDS (ASYNCcnt) path, compute per-lane.

#define WINDOW 6
#define DIM 32
#define PAD_DW 196          // 192 data dwords + 4 pad => conflict-free b128 LDS reads
#define THREADS 64          // 2 waves/block; static LDS = 64*196*4 = 50176 B
#define ROW_BYTES (WINDOW * DIM * 4)  // 768

// one 16B async chunk; INST_OFFSET is added to BOTH the LDS and global address
#define ASY(off) "global_load_async_to_lds_b128 %0, %1, off offset:" #off "\n\t"

__global__ __launch_bounds__(THREADS) void evo_bank_kernel(
    const float* __restrict__ bank,
    const float* __restrict__ emb,
    const int*   __restrict__ idx,
    const int*   __restrict__ ptr,
    float* __restrict__ out,
    int B)
{
    __shared__ __align__(16) float smem[THREADS * PAD_DW];

    const int tid = threadIdx.x;
    int b = blockIdx.x * THREADS + tid;
    const bool valid = (b < B);
    const int bc = valid ? b : (B - 1);       // clamp: keep all 32 lanes active

    const int row = idx[bc];                  // gather row index
    const unsigned p = ((unsigned)ptr[row]) % WINDOW;

    // ---- async-stage this lane's 768B bank row into its padded LDS region ----
    unsigned long long gsrc =
        (unsigned long long)(const void*)bank +
        (unsigned long long)(unsigned)row * (unsigned long long)ROW_BYTES;
    unsigned ldsb = (unsigned)(size_t)(const void*)&smem[tid * PAD_DW];

    asm volatile(
        ASY(0)   ASY(16)  ASY(32)  ASY(48)  ASY(64)  ASY(80)  ASY(96)  ASY(112)
        ASY(128) ASY(144) ASY(160) ASY(176) ASY(192) ASY(208) ASY(224) ASY(240)
        ASY(256) ASY(272) ASY(288) ASY(304) ASY(320) ASY(336) ASY(352) ASY(368)
        ASY(384) ASY(400) ASY(416) ASY(432) ASY(448) ASY(464) ASY(480) ASY(496)
        ASY(512) ASY(528) ASY(544) ASY(560) ASY(576) ASY(592) ASY(608) ASY(624)
        ASY(640) ASY(656) ASY(672) ASY(688) ASY(704) ASY(720) ASY(736) ASY(752)
        :: "v"(ldsb), "v"(gsrc) : "memory");

    // ---- load emb row to registers (overlaps with async traffic) ----
    float4 e[8];
    const float4* ev = (const float4*)(emb + (size_t)bc * DIM);
#pragma unroll
    for (int j = 0; j < 8; ++j) e[j] = ev[j];

    // async engine must finish before we overwrite slot p (scatter-before-gather)
#if __has_builtin(__builtin_amdgcn_s_wait_asynccnt)
    __builtin_amdgcn_s_wait_asynccnt(0);
#else
    asm volatile("s_wait_asynccnt 0x0" ::: "memory");
#endif

    // circular-buffer write: emb replaces window row p for this b
    float4* slot = (float4*)&smem[tid * PAD_DW + p * DIM];
#pragma unroll
    for (int j = 0; j < 8; ++j) slot[j] = e[j];

    // ---- per-lane temporal consistency ----
    const float4* rows = (const float4*)&smem[tid * PAD_DW];
    float inv[WINDOW];
    float dots[WINDOW - 1];
    float4 pr[8];
#pragma unroll
    for (int w = 0; w < WINDOW; ++w) {
        float4 cu[8];
#pragma unroll
        for (int j = 0; j < 8; ++j) cu[j] = rows[w * 8 + j];
        float ss = 0.0f, dt = 0.0f;
#pragma unroll
        for (int j = 0; j < 8; ++j) {
            ss += cu[j].x * cu[j].x + cu[j].y * cu[j].y +
                  cu[j].z * cu[j].z + cu[j].w * cu[j].w;
            if (w > 0)
                dt += pr[j].x * cu[j].x + pr[j].y * cu[j].y +
                      pr[j].z * cu[j].z + pr[j].w * cu[j].w;
        }
        inv[w] = 1.0f / fmaxf(sqrtf(ss), 1e-6f);   // F.normalize(eps=1e-6)
        if (w > 0) dots[w - 1] = dt;
#pragma unroll
        for (int j = 0; j < 8; ++j) pr[j] = cu[j];
    }

    float s[WINDOW - 1];
    float mean = 0.0f;
#pragma unroll
    for (int w = 0; w < WINDOW - 1; ++w) {
        s[w] = dots[w] * inv[w] * inv[w + 1];      // dot of normalized rows
        mean += s[w];
    }
    mean *= (1.0f / (WINDOW - 1));
    float var = 0.0f;
#pragma unroll
    for (int w = 0; w < WINDOW - 1; ++w) {
        float d = s[w] - mean;
        var += d * d;
    }
    var *= (1.0f / (WINDOW - 2));                   // ddof=1 -> /4
    float c = 1.0f / (1.0f + sqrtf(var));
    c = fminf(fmaxf(c, 0.0f), 1.0f);

    if (valid) out[b] = c;
}

extern "C" void kernel_launch(void* const* d_in, const int* in_sizes, int n_in,
                              void* d_out, int out_size, void* d_ws, size_t ws_size,
                              hipStream_t stream) {
    const float* bank = (const float*)d_in[0];   // (1e6, 6, 32) f32
    const float* emb  = (const float*)d_in[1];   // (B, 32)      f32
    const int*   idx  = (const int*)d_in[2];     // (B,)         i32
    const int*   ptr  = (const int*)d_in[3];     // (1e6,)       i32
    // d_in[4] = filled: unused by the output
    float* out = (float*)d_out;                  // (B,) f32

    const int B = in_sizes[2];
    const int blocks = (B + THREADS - 1) / THREADS;
    evo_bank_kernel<<<blocks, THREADS, 0, stream>>>(bank, emb, idx, ptr, out, B);
}